// Weighted_MSE_8770323218587
// MI455X (gfx1250) — compile-verified
//
#include <hip/hip_runtime.h>

// Weighted MSE loss (n=16, c=3, h=512, w=512), fp32 streaming reduction.
// Roofline: ~117.4 MB read once -> ~5 us at 23.3 TB/s; compute negligible.

typedef __attribute__((ext_vector_type(4))) float f4;
typedef __attribute__((ext_vector_type(2))) float v2f;
typedef __attribute__((ext_vector_type(8))) float v8f;

namespace {
constexpr int kN = 16, kC = 3, kH = 512, kW = 512;
constexpr int kHW4    = (kH * kW) / 4;        // 65536 float4 per (n, channel) plane
constexpr int kNOff4  = (kC - 1) * kHW4;      // 131072 extra float4 offset per image
constexpr int kGroups = (kN * kH * kW) / 4;   // 1048576 float4 spatial groups
}

// Full-wave (32-lane) sum via V_WMMA_F32_16X16X4_F32 with an all-ones B matrix.
// A layout (16x4 f32): lanes 0-15 -> (M=lane, K=0/1), lanes 16-31 -> (M=lane-16, K=2/3).
// With a = {v, 0}: A[m][0] = v[m], A[m][2] = v[m+16], rest 0.
// D[m][n] = v[m] + v[m+16] for every column n.
// D layout: VGPR r, lanes 0-15 -> (M=r, N=lane); lanes 16-31 -> (M=r+8, N=lane-16).
// Per-lane sum of d[0..7] gives half the rows of one column; xor-shuffle by 16 completes it.
__device__ __forceinline__ float wave_sum32(float v) {
  v2f a; a[0] = v;    a[1] = 0.0f;
  v2f b; b[0] = 1.0f; b[1] = 1.0f;
  v8f c = {};
  v8f d = __builtin_amdgcn_wmma_f32_16x16x4_f32(
      /*neg_a=*/false, a, /*neg_b=*/false, b,
      /*c_mod=*/(short)0, c, /*reuse_a=*/false, /*reuse_b=*/false);
  float s = d[0] + d[1] + d[2] + d[3] + d[4] + d[5] + d[6] + d[7];
  s += __shfl_xor(s, 16, 32);
  return s;  // every lane holds the full 32-lane sum
}

__global__ void wmse_init(float* __restrict__ acc) {
  acc[0] = 0.0f;  // pos_sum
  acc[1] = 0.0f;  // neg_sum
  acc[2] = 0.0f;  // pos_count
}

__global__ __launch_bounds__(256) void wmse_reduce(const f4* __restrict__ x4,
                                                   const f4* __restrict__ ae4,
                                                   const f4* __restrict__ seg4,
                                                   float* __restrict__ acc) {
  float pos_sum = 0.0f, tot_sum = 0.0f, cnt = 0.0f;
  const int stride = gridDim.x * blockDim.x;
  for (int g = blockIdx.x * blockDim.x + threadIdx.x; g < kGroups; g += stride) {
    const int n    = g >> 16;            // g / kHW4 (kHW4 == 65536)
    const int base = g + n * kNOff4;     // float4 index of channel 0
    // 7 x 128-bit nontemporal loads: seg mask + 3 channels of x and out_ae.
    f4 sv = __builtin_nontemporal_load(&seg4[g]);
    f4 x0 = __builtin_nontemporal_load(&x4[base]);
    f4 x1 = __builtin_nontemporal_load(&x4[base + kHW4]);
    f4 x2 = __builtin_nontemporal_load(&x4[base + 2 * kHW4]);
    f4 a0 = __builtin_nontemporal_load(&ae4[base]);
    f4 a1 = __builtin_nontemporal_load(&ae4[base + kHW4]);
    f4 a2 = __builtin_nontemporal_load(&ae4[base + 2 * kHW4]);
    f4 d0 = a0 - x0, d1 = a1 - x1, d2 = a2 - x2;
    f4 cs = d0 * d0 + d1 * d1 + d2 * d2;   // per-spatial sum over channels
#pragma unroll
    for (int j = 0; j < 4; ++j) {
      const float p = (sv[j] > 0.5f) ? 1.0f : 0.0f;
      cnt     += p;
      pos_sum  = fmaf(p, cs[j], pos_sum);
      tot_sum += cs[j];
    }
  }
  const float neg_sum = tot_sum - pos_sum;

  // Wave32 reduction via WMMA (EXEC is all-1s here: uniform epilogue).
  const float wp = wave_sum32(pos_sum);
  const float wn = wave_sum32(neg_sum);
  const float wc = wave_sum32(cnt);

  __shared__ float lp[8], ln[8], lc[8];   // 256 threads -> 8 wave32
  const int lane = threadIdx.x & 31;
  const int wid  = threadIdx.x >> 5;
  if (lane == 0) { lp[wid] = wp; ln[wid] = wn; lc[wid] = wc; }
  __syncthreads();
  if (threadIdx.x == 0) {
    float p = 0.0f, nn = 0.0f, c = 0.0f;
#pragma unroll
    for (int i = 0; i < 8; ++i) { p += lp[i]; nn += ln[i]; c += lc[i]; }
    atomicAdd(&acc[0], p);
    atomicAdd(&acc[1], nn);
    atomicAdd(&acc[2], c);
  }
}

__global__ void wmse_finalize(const float* __restrict__ acc, float* __restrict__ out) {
  const float total   = (float)(kN * kH * kW);
  const float cf      = (float)kC;
  const float pos_num = acc[2];
  const float neg_num = total - pos_num;
  const float pos_ratio = pos_num / total;
  const float neg_ratio = 1.0f - pos_ratio;
  const float mse_pos = acc[0] / (pos_num * cf);
  const float mse_neg = acc[1] / (neg_num * cf);
  out[0] = pos_ratio * mse_neg + neg_ratio * mse_pos;
}

extern "C" void kernel_launch(void* const* d_in, const int* in_sizes, int n_in,
                              void* d_out, int out_size, void* d_ws, size_t ws_size,
                              hipStream_t stream) {
  (void)in_sizes; (void)n_in; (void)out_size; (void)ws_size;
  const f4* x4   = (const f4*)d_in[0];   // x       (16,3,512,512) f32
  const f4* ae4  = (const f4*)d_in[1];   // out_ae  (16,3,512,512) f32
  const f4* seg4 = (const f4*)d_in[2];   // out_seg (16,1,512,512) f32
  float* acc = (float*)d_ws;             // 3 f32 accumulators
  float* out = (float*)d_out;

  wmse_init<<<1, 1, 0, stream>>>(acc);
  wmse_reduce<<<2048, 256, 0, stream>>>(x4, ae4, seg4, acc);
  wmse_finalize<<<1, 1, 0, stream>>>(acc, out);
}